// GraphDDPMUNet_21638045237663
// MI455X (gfx1250) — compile-verified
//
#include <hip/hip_runtime.h>
#include <cstdint>

#define N0   20000
#define NE   320000
#define HID  256
#define TB   7

typedef __attribute__((ext_vector_type(2))) float v2f;
typedef __attribute__((ext_vector_type(8))) float v8f;
typedef __attribute__((ext_vector_type(4))) unsigned v4u;
typedef __attribute__((ext_vector_type(8))) int v8i;
typedef __attribute__((ext_vector_type(4))) int v4i;

// ---------------------------------------------------------------- helpers
__device__ __forceinline__ float gdu_sig(float x)  { return 1.f / (1.f + __expf(-x)); }
__device__ __forceinline__ float gdu_silu(float x) { return x * gdu_sig(x); }
__device__ __forceinline__ float gdu_wred(float v) {
#pragma unroll
  for (int m = 16; m; m >>= 1) v += __shfl_xor(v, m, 32);
  return v;
}
__device__ __forceinline__ unsigned gdu_key(float f) {
  unsigned u = __float_as_uint(f);
  return (u & 0x80000000u) ? ~u : (u | 0x80000000u);
}

// ---------------------------------------------------------------- TDM: 2D f32 tile -> LDS
// Pads 4 DWORDs after every 256 DWORDs stored => LDS row stride 260 floats.
// OOB rows (beyond tensor_dim1) read as zero (hardware OOB rule), which
// replaces manual zero-fill of the tail A tile.
__device__ __forceinline__ void gdu_tdm_load_2d(const void* gaddr, unsigned lds_off,
                                                unsigned tensor_d0, unsigned tensor_d1,
                                                unsigned tile_d0, unsigned tile_d1,
                                                unsigned stride0) {
  unsigned long long ga = (unsigned long long)(uintptr_t)gaddr;
  v4u g0;
  g0[0] = 1u;                                            // count=1, user descriptor
  g0[1] = lds_off;                                       // lds_addr (bytes)
  g0[2] = (unsigned)(ga & 0xFFFFFFFFu);                  // global_addr[31:0]
  g0[3] = (unsigned)((ga >> 32) & 0x01FFFFFFu)           // global_addr[56:32]
          | (2u << 30);                                  // type = 2 (image)
  v8i g1;
  g1[0] = (int)((2u << 16)        // data_size = 4 bytes
              | (1u << 20)        // pad_enable
              | (7u << 22)        // pad_interval: 256 DWORDs
              | (3u << 25));      // pad_amount: 4 DWORDs
  g1[1] = (int)((tensor_d0 & 0xFFFFu) << 16);            // tensor_dim0[15:0]
  g1[2] = (int)(((tensor_d0 >> 16) & 0xFFFFu) | ((tensor_d1 & 0xFFFFu) << 16));
  g1[3] = (int)(((tensor_d1 >> 16) & 0xFFFFu) | ((tile_d0 & 0xFFFFu) << 16));
  g1[4] = (int)(tile_d1 & 0xFFFFu);                      // tile_dim1; tile_dim2=0 => 2D
  g1[5] = (int)stride0;                                  // tensor_dim0_stride[31:0]
  g1[6] = 0;
  g1[7] = 0;
  v4i gz4 = {0, 0, 0, 0};
  v8i gz8 = {0, 0, 0, 0, 0, 0, 0, 0};
  __builtin_amdgcn_tensor_load_to_lds(g0, g1, gz4, gz4, gz8, 0);
}

// ---------------------------------------------------------------- FiLM MLP (one block)
__global__ void gdu_film(const float* __restrict__ cond,
                         const float* __restrict__ w1, const float* __restrict__ b1,
                         const float* __restrict__ w2, const float* __restrict__ b2,
                         const float* __restrict__ w3, const float* __restrict__ b3,
                         float* __restrict__ gamma, float* __restrict__ beta) {
  __shared__ float cs[128];
  __shared__ float h1[512];
  __shared__ float h2[512];
  int t = threadIdx.x;
  if (t < 128) cs[t] = cond[t];
  __syncthreads();
  for (int u = t; u < 512; u += 256) {
    float s = b1[u];
    for (int k = 0; k < 128; ++k) s += w1[u * 128 + k] * cs[k];
    h1[u] = gdu_silu(s);
  }
  __syncthreads();
  for (int u = t; u < 512; u += 256) {
    float s = b2[u];
    for (int k = 0; k < 512; ++k) s += w2[u * 512 + k] * h1[k];
    h2[u] = gdu_silu(s);
  }
  __syncthreads();
  for (int u = t; u < TB * 512; u += 256) {
    float s = b3[u];
    for (int k = 0; k < 512; ++k) s += w3[u * 512 + k] * h2[k];
    int blk = u >> 9, within = u & 511;
    if (within < 256) gamma[blk * 256 + within] = 1.f + s;
    else              beta[blk * 256 + (within - 256)] = s;
  }
}

// ---------------------------------------------------------------- input + positional projection
__global__ void gdu_proj(const float* __restrict__ x0, const float* __restrict__ pos,
                         const float* __restrict__ in_w, const float* __restrict__ in_b,
                         const float* __restrict__ pw1, const float* __restrict__ pb1,
                         float* __restrict__ h0, float* __restrict__ tpos) {
  int tid = blockIdx.x * 256 + threadIdx.x;
  int i = tid >> 8, c = tid & 255;
  float a = in_b[c], p = pb1[c];
#pragma unroll
  for (int k = 0; k < 3; ++k) {
    a += x0[i * 3 + k] * in_w[c * 3 + k];
    p += pos[i * 3 + k] * pw1[c * 3 + k];
  }
  h0[tid] = a;
  tpos[tid] = gdu_silu(p);
}

// ---------------------------------------------------------------- WMMA fp32 GEMM: out = A[n,256] @ W[256,256]^T (+epilogue)
// mode 0: out = resid + (acc + bias)
// mode 1: out = resid + (acc + bias)*gamma + beta     (FiLM GCN block)
#define GEMM_MROWS 32
#define GEMM_LDS_FLOATS (GEMM_MROWS * 260 + 256 * 260)

__global__ void gdu_gemm_wmma(const float* __restrict__ A, const float* __restrict__ W,
                              const float* __restrict__ bias, const float* __restrict__ resid,
                              const float* __restrict__ gamma, const float* __restrict__ beta,
                              float* __restrict__ out, int n, int mode) {
  extern __shared__ float sm[];
  float* As = sm;                       // 32 x 260 (padded via TDM)
  float* Ws = sm + GEMM_MROWS * 260;    // 256 x 260 (padded via TDM) — CDNA5 320KB LDS
  const int t  = threadIdx.x;
  const int r0 = blockIdx.x * GEMM_MROWS;
  const int wv = t >> 5;

  // Tensor Data Mover staging: wave 0 pulls W, wave 1 pulls the A tile.
  if (wv == 0) {
    gdu_tdm_load_2d(W, (unsigned)(uintptr_t)(void*)Ws, 256u, 256u, 256u, 256u, 256u);
    __builtin_amdgcn_s_wait_tensorcnt(0);
  } else if (wv == 1) {
    gdu_tdm_load_2d(A + (size_t)r0 * HID, (unsigned)(uintptr_t)(void*)As,
                    256u, (unsigned)(n - r0), 256u, (unsigned)GEMM_MROWS, 256u);
    __builtin_amdgcn_s_wait_tensorcnt(0);
  }
  __syncthreads();

  const int lane   = t & 31;
  const int mt     = wv & 1;            // M tile (0..1)
  const int ntBase = (wv >> 1) * 4;     // 4 N tiles per wave
  const int rr     = lane & 15;
  const int khalf  = (lane >> 4) * 2;   // A/B fragment K-pair select

  const float* Ap = As + (mt * 16 + rr) * 260;
  const float* Bp[4];
#pragma unroll
  for (int j = 0; j < 4; ++j) Bp[j] = Ws + ((ntBase + j) * 16 + rr) * 260;

  v8f acc[4];
  v8f zero8 = {0.f, 0.f, 0.f, 0.f, 0.f, 0.f, 0.f, 0.f};
#pragma unroll
  for (int j = 0; j < 4; ++j) acc[j] = zero8;

  // Batch the 5 LDS fragment loads per K-step, then issue 4 WMMAs back-to-back:
  // one s_wait_dscnt feeding 4 matrix ops instead of a wait per WMMA.
#pragma unroll 2
  for (int k0 = 0; k0 < HID; k0 += 4) {
    v2f a;
    a.x = Ap[k0 + khalf];
    a.y = Ap[k0 + khalf + 1];
    v2f bf[4];
#pragma unroll
    for (int j = 0; j < 4; ++j) {
      bf[j].x = Bp[j][k0 + khalf];
      bf[j].y = Bp[j][k0 + khalf + 1];
    }
#pragma unroll
    for (int j = 0; j < 4; ++j)
      acc[j] = __builtin_amdgcn_wmma_f32_16x16x4_f32(false, a, false, bf[j], (short)0,
                                                     acc[j], false, false);
  }

  const int mbase = r0 + mt * 16 + 8 * (lane >> 4);
#pragma unroll
  for (int j = 0; j < 4; ++j) {
    int gc = (ntBase + j) * 16 + (lane & 15);
#pragma unroll
    for (int i = 0; i < 8; ++i) {
      int gr = mbase + i;
      if (gr < n) {
        float v = acc[j][i] + bias[gc];
        float res = resid[(size_t)gr * HID + gc];
        float o;
        if (mode == 0) o = res + v;
        else           o = res + v * gamma[gc] + beta[gc];
        out[(size_t)gr * HID + gc] = o;
      }
    }
  }
}

// ---------------------------------------------------------------- LayerNorm + SiLU (wave per row)
__global__ void gdu_ln_silu(const float* __restrict__ x, const float* __restrict__ g,
                            const float* __restrict__ b, float* __restrict__ y, int n) {
  int wave = threadIdx.x >> 5, lane = threadIdx.x & 31;
  int row = blockIdx.x * 8 + wave;
  if (row >= n) return;  // uniform per wave; kernel has no barriers
  const float* xr = x + (size_t)row * HID;
  float vals[8], s = 0.f, s2 = 0.f;
#pragma unroll
  for (int j = 0; j < 8; ++j) {
    float v = xr[lane + 32 * j];
    vals[j] = v; s += v; s2 += v * v;
  }
  s = gdu_wred(s); s2 = gdu_wred(s2);
  float mean = s * (1.f / HID);
  float var  = s2 * (1.f / HID) - mean * mean;
  float rstd = rsqrtf(var + 1e-5f);
#pragma unroll
  for (int j = 0; j < 8; ++j) {
    int c = lane + 32 * j;
    float t = (vals[j] - mean) * rstd * g[c] + b[c];
    y[(size_t)row * HID + c] = gdu_silu(t);
  }
}

// ---------------------------------------------------------------- SpMM scatter (edge x 8 channels per thread)
__global__ void gdu_spmm_scatter(const int* __restrict__ rw, const int* __restrict__ cl,
                                 const float* __restrict__ val, const float* __restrict__ hh,
                                 float* __restrict__ z, int ecnt) {
  int gid = blockIdx.x * 256 + threadIdx.x;
  int e = gid >> 5;
  if (e >= ecnt) return;
  float v = val[e];
  if (v == 0.f) return;
  int c0 = (gid & 31) * 8;
  const float* s = hh + (size_t)cl[e] * HID + c0;
  float* d = z + (size_t)rw[e] * HID + c0;
#pragma unroll
  for (int i = 0; i < 8; ++i) atomicAdd(d + i, v * s[i]);
}

__global__ void gdu_diag_axpy(float* __restrict__ z, const float* __restrict__ diag,
                              const float* __restrict__ hh) {
  int tid = blockIdx.x * 256 + threadIdx.x;
  int i = tid >> 8;
  z[tid] += diag[i] * hh[tid];
}

// ---------------------------------------------------------------- GCN normalization
__global__ void gdu_deg_init(float* __restrict__ deg, const float* __restrict__ din, int n) {
  int i = blockIdx.x * 256 + threadIdx.x;
  if (i < n) deg[i] = (din ? din[i] : 0.f) + 1.f;
}
__global__ void gdu_deg_edge(const int* __restrict__ rw, const float* __restrict__ vin,
                             float* __restrict__ deg, int ecnt) {
  int e = blockIdx.x * 256 + threadIdx.x;
  if (e < ecnt) atomicAdd(&deg[rw[e]], vin ? vin[e] : 1.f);
}
__global__ void gdu_norm_fin(const float* __restrict__ deg, const float* __restrict__ din,
                             float* __restrict__ dinv, float* __restrict__ dout, int n) {
  int i = blockIdx.x * 256 + threadIdx.x;
  if (i < n) {
    float di = rsqrtf(deg[i]);
    dinv[i] = di;
    dout[i] = ((din ? din[i] : 0.f) + 1.f) * di * di;
  }
}
__global__ void gdu_norm_edges(const int* __restrict__ rw, const int* __restrict__ cl,
                               const float* __restrict__ vin, const float* __restrict__ dinv,
                               float* __restrict__ vout, int ecnt) {
  int e = blockIdx.x * 256 + threadIdx.x;
  if (e < ecnt) vout[e] = (vin ? vin[e] : 1.f) * dinv[rw[e]] * dinv[cl[e]];
}

// ---------------------------------------------------------------- pooling scores (wave per row)
__global__ void gdu_pool_score(const float* __restrict__ h, const float* __restrict__ g,
                               const float* __restrict__ b, const float* __restrict__ w1,
                               const float* __restrict__ b1, const float* __restrict__ w2,
                               const float* __restrict__ b2p, float* __restrict__ score,
                               unsigned* __restrict__ keys, int n) {
  __shared__ float xn[8][260];
  int wave = threadIdx.x >> 5, lane = threadIdx.x & 31;
  int row = blockIdx.x * 8 + wave;  // n divisible by 8 at all call sites
  const float* xr = h + (size_t)row * HID;
  float vals[8], s = 0.f, s2 = 0.f;
#pragma unroll
  for (int j = 0; j < 8; ++j) {
    float v = xr[lane + 32 * j];
    vals[j] = v; s += v; s2 += v * v;
  }
  s = gdu_wred(s); s2 = gdu_wred(s2);
  float mean = s * (1.f / HID);
  float rstd = rsqrtf(s2 * (1.f / HID) - mean * mean + 1e-5f);
#pragma unroll
  for (int j = 0; j < 8; ++j) {
    int c = lane + 32 * j;
    xn[wave][c] = (vals[j] - mean) * rstd * g[c] + b[c];
  }
  __syncthreads();
  float acc = 0.f;
  for (int u = lane; u < 128; u += 32) {
    float d = b1[u];
    for (int k = 0; k < HID; ++k) d += w1[u * HID + k] * xn[wave][k];
    acc += gdu_silu(d) * w2[u];
  }
  acc = gdu_wred(acc);
  if (lane == 0) {
    float sc = acc + b2p[0];
    score[row] = sc;
    keys[row] = gdu_key(sc);
  }
}

// ---------------------------------------------------------------- top-k radix select
__global__ void gdu_radix_init(int* state, int k) {
  state[0] = 0; state[1] = k;
}
__global__ void gdu_hist(const unsigned* __restrict__ keys, int n, const int* __restrict__ state,
                         int shift, unsigned* __restrict__ hist) {
  __shared__ unsigned lh[256];
  int t = threadIdx.x;
  lh[t] = 0;
  __syncthreads();
  unsigned prefix = (unsigned)state[0];
  unsigned maskAbove = (shift >= 24) ? 0u : (0xFFFFFFFFu << (shift + 8));
  for (int i = blockIdx.x * 256 + t; i < n; i += gridDim.x * 256) {
    unsigned kk = keys[i];
    if ((kk & maskAbove) == (prefix & maskAbove))
      atomicAdd(&lh[(kk >> shift) & 255u], 1u);
  }
  __syncthreads();
  if (lh[t]) atomicAdd(&hist[t], lh[t]);
}
__global__ void gdu_digit(const unsigned* __restrict__ hist, int shift, int* state) {
  unsigned prefix = (unsigned)state[0];
  int rem = state[1], cum = 0;
  for (int d = 255; d >= 0; --d) {
    int c = (int)hist[d];
    if (cum + c >= rem) {
      state[0] = (int)(prefix | ((unsigned)d << shift));
      state[1] = rem - cum;
      return;
    }
    cum += c;
  }
}
// deterministic compaction: ascending-index order among selected
__global__ void gdu_compact(const unsigned* __restrict__ keys, const int* __restrict__ state,
                            int n, int* __restrict__ keep, int* __restrict__ mark,
                            int* __restrict__ inv) {
  __shared__ int sg[256], sq[256];
  unsigned kth = (unsigned)state[0];
  int eq_need = state[1];
  int t = threadIdx.x;
  int chunk = (n + 255) >> 8;
  int lo = t * chunk, hi = min(n, lo + chunk);
  int cg = 0, cq = 0;
  for (int i = lo; i < hi; ++i) {
    unsigned kk = keys[i];
    cg += (kk > kth);
    cq += (kk == kth);
  }
  sg[t] = cg; sq[t] = cq;
  __syncthreads();
  if (t == 0) {
    int ag = 0, aq = 0;
    for (int i = 0; i < 256; ++i) {
      int g0 = sg[i], q0 = sq[i];
      sg[i] = ag; sq[i] = aq;
      ag += g0; aq += q0;
    }
  }
  __syncthreads();
  int g = sg[t], q = sq[t];
  for (int i = lo; i < hi; ++i) {
    unsigned kk = keys[i];
    if (kk > kth) {
      int p = g + min(q, eq_need);
      keep[p] = i; mark[i] = 1; inv[i] = p; g++;
    } else if (kk == kth) {
      if (q < eq_need) {
        int p = g + q;
        keep[p] = i; mark[i] = 1; inv[i] = p;
      }
      q++;
    }
  }
}

// ---------------------------------------------------------------- pooled graph rebuild / gathers
__global__ void gdu_pool_edges(const int* __restrict__ r, const int* __restrict__ c,
                               const float* __restrict__ v, const int* __restrict__ mark,
                               const int* __restrict__ inv, int* __restrict__ r2,
                               int* __restrict__ c2, float* __restrict__ v2, int ecnt) {
  int e = blockIdx.x * 256 + threadIdx.x;
  if (e >= ecnt) return;
  int a = r[e], bb = c[e];
  bool ok = mark[a] && mark[bb];
  r2[e] = ok ? inv[a] : 0;
  c2[e] = ok ? inv[bb] : 0;
  v2[e] = ok ? v[e] : 0.f;
}
__global__ void gdu_gather_diag(const float* __restrict__ dprev, const int* __restrict__ keep,
                                float* __restrict__ dout, int k) {
  int j = blockIdx.x * 256 + threadIdx.x;
  if (j < k) dout[j] = dprev[keep[j]];
}
__global__ void gdu_gather_rows(const float* __restrict__ in, const int* __restrict__ keep,
                                float* __restrict__ out) {
  int tid = blockIdx.x * 256 + threadIdx.x;
  int j = tid >> 8, c = tid & 255;
  out[tid] = in[(size_t)keep[j] * HID + c];
}
__global__ void gdu_unpool(const float* __restrict__ skip, const float* __restrict__ small,
                           const int* __restrict__ mark, const int* __restrict__ inv,
                           float* __restrict__ out) {
  int tid = blockIdx.x * 256 + threadIdx.x;
  int i = tid >> 8, c = tid & 255;
  float v = skip[tid];
  if (mark[i]) v += small[(size_t)inv[i] * HID + c];
  out[tid] = v;
}

// ---------------------------------------------------------------- output head (wave per row)
__global__ void gdu_out_head(const float* __restrict__ h, const float* __restrict__ g,
                             const float* __restrict__ b, const float* __restrict__ ow,
                             const float* __restrict__ ob, float* __restrict__ out, int n) {
  int wave = threadIdx.x >> 5, lane = threadIdx.x & 31;
  int row = blockIdx.x * 8 + wave;  // n divisible by 8
  const float* xr = h + (size_t)row * HID;
  float vals[8], s = 0.f, s2 = 0.f;
#pragma unroll
  for (int j = 0; j < 8; ++j) {
    float v = xr[lane + 32 * j];
    vals[j] = v; s += v; s2 += v * v;
  }
  s = gdu_wred(s); s2 = gdu_wred(s2);
  float mean = s * (1.f / HID);
  float rstd = rsqrtf(s2 * (1.f / HID) - mean * mean + 1e-5f);
  float y[8];
#pragma unroll
  for (int j = 0; j < 8; ++j) {
    int c = lane + 32 * j;
    y[j] = gdu_silu((vals[j] - mean) * rstd * g[c] + b[c]);
  }
#pragma unroll
  for (int o = 0; o < 3; ++o) {
    float p = 0.f;
#pragma unroll
    for (int j = 0; j < 8; ++j) p += y[j] * ow[o * HID + lane + 32 * j];
    p = gdu_wred(p);
    if (lane == 0) out[row * 3 + o] = p + ob[o];
  }
}

// ================================================================ host orchestration
extern "C" void kernel_launch(void* const* d_in, const int* in_sizes, int n_in,
                              void* d_out, int out_size, void* d_ws, size_t ws_size,
                              hipStream_t stream) {
  (void)in_sizes; (void)n_in; (void)out_size; (void)ws_size;
  const float* x0      = (const float*)d_in[0];
  const float* pos     = (const float*)d_in[1];
  const float* cond    = (const float*)d_in[2];
  const int*   srcE    = (const int*)d_in[3];
  const int*   dstE    = (const int*)d_in[4];
  const float* in_w    = (const float*)d_in[5];
  const float* in_b    = (const float*)d_in[6];
  const float* pos_w1  = (const float*)d_in[7];
  const float* pos_b1  = (const float*)d_in[8];
  const float* pos_w2  = (const float*)d_in[9];
  const float* pos_b2  = (const float*)d_in[10];
  const float* ln_g    = (const float*)d_in[11];
  const float* ln_b    = (const float*)d_in[12];
  const float* conv_w  = (const float*)d_in[13];
  const float* conv_b  = (const float*)d_in[14];
  const float* pool_ln_g = (const float*)d_in[15];
  const float* pool_ln_b = (const float*)d_in[16];
  const float* pool_w1 = (const float*)d_in[17];
  const float* pool_b1 = (const float*)d_in[18];
  const float* pool_w2 = (const float*)d_in[19];
  const float* pool_b2 = (const float*)d_in[20];
  const float* film_w1 = (const float*)d_in[21];
  const float* film_b1 = (const float*)d_in[22];
  const float* film_w2 = (const float*)d_in[23];
  const float* film_b2 = (const float*)d_in[24];
  const float* film_w3 = (const float*)d_in[25];
  const float* film_b3 = (const float*)d_in[26];
  const float* out_g   = (const float*)d_in[27];
  const float* out_b   = (const float*)d_in[28];
  const float* out_w   = (const float*)d_in[29];
  const float* out_bias= (const float*)d_in[30];
  float* result = (float*)d_out;

  const int ns[4] = {20000, 10000, 5000, 2500};

  char* wp = (char*)d_ws;
  auto alloc = [&](size_t bytes) -> char* {
    uintptr_t a = ((uintptr_t)wp + 255) & ~(uintptr_t)255;
    wp = (char*)(a + bytes);
    return (char*)a;
  };

  float* gamma = (float*)alloc((size_t)TB * HID * 4);
  float* beta  = (float*)alloc((size_t)TB * HID * 4);
  float* hh    = (float*)alloc((size_t)N0 * HID * 4);
  float* zb    = (float*)alloc((size_t)N0 * HID * 4);
  float* bufA  = (float*)alloc((size_t)N0 * HID * 4);
  float* bufB  = (float*)alloc((size_t)N0 * HID * 4);
  float* bufC  = (float*)alloc((size_t)ns[1] * HID * 4);
  float* bufD  = (float*)alloc((size_t)ns[1] * HID * 4);
  float* bufE  = (float*)alloc((size_t)ns[2] * HID * 4);
  float* bufF  = (float*)alloc((size_t)ns[2] * HID * 4);
  float* bufG  = (float*)alloc((size_t)ns[3] * HID * 4);
  float* bufH  = (float*)alloc((size_t)ns[3] * HID * 4);
  float* bufI  = (float*)alloc((size_t)ns[1] * HID * 4);

  int* rowW[4] = {nullptr, nullptr, nullptr, nullptr};
  int* colW[4] = {nullptr, nullptr, nullptr, nullptr};
  for (int l = 1; l < 4; ++l) {
    rowW[l] = (int*)alloc((size_t)NE * 4);
    colW[l] = (int*)alloc((size_t)NE * 4);
  }
  float* valL[4]; float* diagL[4];
  for (int l = 0; l < 4; ++l) {
    valL[l]  = (float*)alloc((size_t)NE * 4);
    diagL[l] = (float*)alloc((size_t)ns[l] * 4);
  }
  const int* rowL[4] = {srcE, rowW[1], rowW[2], rowW[3]};
  const int* colL[4] = {dstE, colW[1], colW[2], colW[3]};

  float* deg    = (float*)alloc((size_t)N0 * 4);
  float* dinv   = (float*)alloc((size_t)N0 * 4);
  float* dtmp   = (float*)alloc((size_t)ns[1] * 4);
  float* scores = (float*)alloc((size_t)N0 * 4);
  unsigned* keys = (unsigned*)alloc((size_t)N0 * 4);
  int* keepL[3]; int* markL[3]; int* invL[3];
  for (int d = 0; d < 3; ++d) {
    keepL[d] = (int*)alloc((size_t)ns[d + 1] * 4);
    markL[d] = (int*)alloc((size_t)ns[d] * 4);
    invL[d]  = (int*)alloc((size_t)ns[d] * 4);
  }
  unsigned* hist = (unsigned*)alloc(256 * 4);
  int* state = (int*)alloc(64);

  constexpr size_t GLDS = (size_t)GEMM_LDS_FLOATS * sizeof(float);

  auto gcn = [&](const int* rw, const int* cl, const float* vin, const float* din,
                 float* vout, float* dout, int n) {
    gdu_deg_init<<<(n + 255) / 256, 256, 0, stream>>>(deg, din, n);
    gdu_deg_edge<<<NE / 256, 256, 0, stream>>>(rw, vin, deg, NE);
    gdu_norm_fin<<<(n + 255) / 256, 256, 0, stream>>>(deg, din, dinv, dout, n);
    gdu_norm_edges<<<NE / 256, 256, 0, stream>>>(rw, cl, vin, dinv, vout, NE);
  };

  auto run_block = [&](const float* hin, float* hout, int blk, int n,
                       const int* rw, const int* cl, const float* vl, const float* dg) {
    gdu_ln_silu<<<(n + 7) / 8, 256, 0, stream>>>(hin, ln_g + blk * HID, ln_b + blk * HID, hh, n);
    (void)hipMemsetAsync(zb, 0, (size_t)n * HID * 4, stream);
    gdu_spmm_scatter<<<(NE * 32) / 256, 256, 0, stream>>>(rw, cl, vl, hh, zb, NE);
    gdu_diag_axpy<<<n, 256, 0, stream>>>(zb, dg, hh);
    gdu_gemm_wmma<<<(n + GEMM_MROWS - 1) / GEMM_MROWS, 256, GLDS, stream>>>(
        zb, conv_w + (size_t)blk * HID * HID, conv_b + blk * HID, hin,
        gamma + blk * HID, beta + blk * HID, hout, n, 1);
  };

  auto do_pool = [&](int d, const float* hpost, float* hsmall) {
    int n = ns[d], k = ns[d + 1];
    gdu_pool_score<<<n / 8, 256, 0, stream>>>(hpost, pool_ln_g + d * HID, pool_ln_b + d * HID,
        pool_w1 + (size_t)d * 128 * HID, pool_b1 + d * 128, pool_w2 + d * 128, pool_b2 + d,
        scores, keys, n);
    gdu_radix_init<<<1, 1, 0, stream>>>(state, k);
    const int shifts[4] = {24, 16, 8, 0};
    for (int pI = 0; pI < 4; ++pI) {
      (void)hipMemsetAsync(hist, 0, 256 * 4, stream);
      gdu_hist<<<80, 256, 0, stream>>>(keys, n, state, shifts[pI], hist);
      gdu_digit<<<1, 1, 0, stream>>>(hist, shifts[pI], state);
    }
    (void)hipMemsetAsync(markL[d], 0, (size_t)n * 4, stream);
    (void)hipMemsetAsync(invL[d], 0, (size_t)n * 4, stream);
    gdu_compact<<<1, 256, 0, stream>>>(keys, state, n, keepL[d], markL[d], invL[d]);
    gdu_pool_edges<<<NE / 256, 256, 0, stream>>>(rowL[d], colL[d], valL[d], markL[d], invL[d],
        rowW[d + 1], colW[d + 1], valL[d + 1], NE);
    gdu_gather_diag<<<(k + 255) / 256, 256, 0, stream>>>(diagL[d], keepL[d], dtmp, k);
    gcn(rowL[d + 1], colL[d + 1], valL[d + 1], dtmp, valL[d + 1], diagL[d + 1], k);
    gdu_gather_rows<<<k, 256, 0, stream>>>(hpost, keepL[d], hsmall);
  };

  // ---- FiLM conditioning
  gdu_film<<<1, 256, 0, stream>>>(cond, film_w1, film_b1, film_w2, film_b2, film_w3, film_b3,
                                  gamma, beta);
  // ---- encode: h = x0@in_w.T + in_b + silu(pos@pos_w1.T+b1)@pos_w2.T + pos_b2
  gdu_proj<<<N0, 256, 0, stream>>>(x0, pos, in_w, in_b, pos_w1, pos_b1, zb, hh);
  gdu_gemm_wmma<<<(N0 + GEMM_MROWS - 1) / GEMM_MROWS, 256, GLDS, stream>>>(
      hh, pos_w2, pos_b2, zb, nullptr, nullptr, bufA, N0, 0);
  // ---- level-0 adjacency normalization
  gcn(rowL[0], colL[0], nullptr, nullptr, valL[0], diagL[0], ns[0]);

  // ---- down path
  run_block(bufA, bufB, 0, ns[0], rowL[0], colL[0], valL[0], diagL[0]);
  do_pool(0, bufB, bufC);
  run_block(bufC, bufD, 1, ns[1], rowL[1], colL[1], valL[1], diagL[1]);
  do_pool(1, bufD, bufE);
  run_block(bufE, bufF, 2, ns[2], rowL[2], colL[2], valL[2], diagL[2]);
  do_pool(2, bufF, bufG);
  // ---- bottleneck
  run_block(bufG, bufH, 3, ns[3], rowL[3], colL[3], valL[3], diagL[3]);
  // ---- up path
  gdu_unpool<<<ns[2], 256, 0, stream>>>(bufF, bufH, markL[2], invL[2], bufE);
  run_block(bufE, bufA, 4, ns[2], rowL[2], colL[2], valL[2], diagL[2]);
  gdu_unpool<<<ns[1], 256, 0, stream>>>(bufD, bufA, markL[1], invL[1], bufC);
  run_block(bufC, bufI, 5, ns[1], rowL[1], colL[1], valL[1], diagL[1]);
  gdu_unpool<<<ns[0], 256, 0, stream>>>(bufB, bufI, markL[0], invL[0], bufA);
  run_block(bufA, bufB, 6, ns[0], rowL[0], colL[0], valL[0], diagL[0]);
  // ---- output head
  gdu_out_head<<<ns[0] / 8, 256, 0, stream>>>(bufB, out_g, out_b, out_w, out_bias, result, ns[0]);
}